// DiscriminatorIndependent_38173669327355
// MI455X (gfx1250) — compile-verified
//
#include <hip/hip_runtime.h>

typedef __attribute__((ext_vector_type(2))) float v2f;
typedef __attribute__((ext_vector_type(8))) float v8f;

#define AS1 __attribute__((address_space(1)))
#define AS3 __attribute__((address_space(3)))

#define N_MLP   1024
#define TILE_B  16
// LDS layout in dwords
#define XS_OFF  0            // x transposed:  [1024][17] (16 used + 1 pad)
#define XA_OFF  17408        // x_aug transposed
#define WP_OFF  34816        // packed weights: [1024][4] float4 {w1a,w1b,b1,w2}
#define RA_OFF  51200        // 16 row accumulators
#define BS_OFF  51216        // bias sum
#define LDS_DWORDS 51232

#if defined(__AMDGCN__) && __has_builtin(__builtin_amdgcn_global_load_async_to_lds_b32)
#define HAVE_ASYNC_LDS 1
#else
#define HAVE_ASYNC_LDS 0
#endif

__global__ __launch_bounds__(256) void mlp1024_wmma_kernel(
    const float* __restrict__ x,  const float* __restrict__ xaug,
    const float* __restrict__ W1, const float* __restrict__ b1,
    const float* __restrict__ W2, const float* __restrict__ b2,
    float* __restrict__ out)
{
  extern __shared__ float lds[];
  float* ldsx   = lds + XS_OFF;
  float* ldsa   = lds + XA_OFF;
  float* wp     = lds + WP_OFF;
  float* rowAcc = lds + RA_OFF;

  const int tid = threadIdx.x;
  const int rowbase = blockIdx.x * TILE_B;

  if (tid < 17) lds[RA_OFF + tid] = 0.0f;   // rowAcc[16] + biasAcc

  // ---- Stage x / x_aug tile transposed into LDS: ldsx[n*17 + m] = x[(rowbase+m)*1024 + n]
#if HAVE_ASYNC_LDS
  // CDNA5 async DMA: per-lane global src (coalesced) -> per-lane LDS dst (stride-17
  // transposed scatter; 17 coprime 64 -> conflict-free banks). No VGPR round-trip,
  // tracked by ASYNCcnt.
  for (int i = 0; i < 64; ++i) {              // 16384 elements per array
    int idx = i * 256 + tid;
    int n = idx & 1023;                       // consecutive lanes -> consecutive n
    int m = idx >> 10;
    const float* gx = x    + ((rowbase + m) << 10) + n;
    const float* ga = xaug + ((rowbase + m) << 10) + n;
    float* lx = ldsx + (n * 17 + m);
    float* la = ldsa + (n * 17 + m);
    __builtin_amdgcn_global_load_async_to_lds_b32((AS1 int*)gx, (AS3 int*)lx, 0, 0);
    __builtin_amdgcn_global_load_async_to_lds_b32((AS1 int*)ga, (AS3 int*)la, 0, 0);
  }
#else
  const float4* xg4 = (const float4*)x;
  const float4* ag4 = (const float4*)xaug;
  #pragma unroll 4
  for (int i = 0; i < 16; ++i) {              // 4096 float4 per array
    int idx4 = i * 256 + tid;
    int n4 = idx4 & 255;
    int m  = idx4 >> 8;
    float4 vx = xg4[(rowbase + m) * 256 + n4];
    float4 va = ag4[(rowbase + m) * 256 + n4];
    int base = (n4 * 4) * 17 + m;
    ldsx[base     ] = vx.x; ldsx[base + 17] = vx.y;
    ldsx[base + 34] = vx.z; ldsx[base + 51] = vx.w;
    ldsa[base     ] = va.x; ldsa[base + 17] = va.y;
    ldsa[base + 34] = va.z; ldsa[base + 51] = va.w;
  }
#endif

  // ---- Stage packed weights: wp[n*16 + j*4 + {0..3}] = {W1[n,j,0], W1[n,j,1], b1[n,j], W2[n,j]}
  #pragma unroll 4
  for (int i = 0; i < 16; ++i) {              // 4096 (n,j) items
    int item = i * 256 + tid;
    int n = item >> 2, j = item & 3;
    int o = n * 16 + j * 4;
    wp[o + 0] = W1[n * 8 + j * 2 + 0];
    wp[o + 1] = W1[n * 8 + j * 2 + 1];
    wp[o + 2] = b1[n * 4 + j];
    wp[o + 3] = W2[n * 4 + j];
  }

  // Per-thread partial of sum_n b2[n] (row-independent constant)
  float bsum = b2[tid] + b2[tid + 256] + b2[tid + 512] + b2[tid + 768];

#if HAVE_ASYNC_LDS
  asm volatile("s_wait_asynccnt 0x0" ::: "memory");  // this wave's DMA done
#endif
  __syncthreads();                                   // all waves' tiles visible
  atomicAdd(&lds[BS_OFF], bsum);

  // ---- WMMA main loop: D(16x16) = A(16x4) * B(4x16) + C, two MLPs per WMMA (block-diagonal K)
  const int lane = tid & 31;
  const int wave = __builtin_amdgcn_readfirstlane(tid >> 5); // scalar -> uniform loop bounds
  const int half = lane >> 4;                 // K/V half (ISA: lanes 16-31 hold K=2,3 / M+8)
  const int c    = lane & 15;                 // A: M row; B/C/D: N column
  const int j    = c & 3;                     // hidden-unit index within a column group
  const bool colOK  = (c < 8);                // columns 8..15 are zero padding
  const int  csel   = colOK ? (c >> 2) : 0;   // which MLP of the pair this column belongs to
  const bool active = colOK && ((c >> 2) == half); // lane carries nonzero B element

  v8f acc = {0.f, 0.f, 0.f, 0.f, 0.f, 0.f, 0.f, 0.f};
  const int p0 = wave * 64;                   // 64 MLP-pairs per wave (8 waves * 128 MLPs)
  for (int p = p0; p < p0 + 64; ++p) {
    int na = 2 * p + half;                    // n for this lane's A elements
    int nc = 2 * p + csel;                    // n owning this lane's column
    float a0 = ldsx[na * 17 + c];             // A[m, 2*half]   = x
    float a1 = ldsa[na * 17 + c];             // A[m, 2*half+1] = x_aug
    const float4 wv = *(const float4*)&wp[nc * 16 + j * 4];
    float b0v = active ? wv.x : 0.0f;         // B[2*half  , c]
    float b1v = active ? wv.y : 0.0f;         // B[2*half+1, c]
    float cb  = colOK  ? wv.z : 0.0f;         // column bias b1
    float w2c = colOK  ? wv.w : 0.0f;         // column W2 coefficient
    v2f A  = {a0, a1};
    v2f Bv = {b0v, b1v};
    v8f C  = {cb, cb, cb, cb, cb, cb, cb, cb};
    v8f D = __builtin_amdgcn_wmma_f32_16x16x4_f32(
        /*neg_a=*/false, A, /*neg_b=*/false, Bv,
        /*c_mod=*/(short)0, C, /*reuse_a=*/false, /*reuse_b=*/false);
    #pragma unroll
    for (int v = 0; v < 8; ++v) {
      float h = fmaxf(D[v], 0.0f);            // relu
      acc[v] = fmaf(h, w2c, acc[v]);          // second layer + reduce over columns
    }
  }

  // ---- Reduce across the 16 lanes of each half (rows 0-7 in lanes 0-15, rows 8-15 in 16-31)
  #pragma unroll
  for (int v = 0; v < 8; ++v) {
    float s = acc[v];
    s += __shfl_xor(s, 1, 32);
    s += __shfl_xor(s, 2, 32);
    s += __shfl_xor(s, 4, 32);
    s += __shfl_xor(s, 8, 32);
    acc[v] = s;
  }
  if (c == 0) {                               // lane 0 (rows 0-7) and lane 16 (rows 8-15)
    #pragma unroll
    for (int v = 0; v < 8; ++v)
      atomicAdd(&rowAcc[v + 8 * half], acc[v]);
  }
  __syncthreads();

  if (tid < TILE_B)
    out[rowbase + tid] = (rowAcc[tid] + lds[BS_OFF]) * (1.0f / 1024.0f);
}

extern "C" void kernel_launch(void* const* d_in, const int* in_sizes, int n_in,
                              void* d_out, int out_size, void* d_ws, size_t ws_size,
                              hipStream_t stream) {
  const float* x    = (const float*)d_in[0];
  const float* xaug = (const float*)d_in[1];
  const float* W1   = (const float*)d_in[2];
  const float* b1   = (const float*)d_in[3];
  const float* W2   = (const float*)d_in[4];
  const float* b2   = (const float*)d_in[5];
  float* out = (float*)d_out;

  const int B = 16384;
  dim3 grid(B / TILE_B);                      // 1024 blocks, each owns 16 batch rows
  dim3 block(256);                            // 8 wave32s
  size_t shmem = (size_t)LDS_DWORDS * sizeof(float);  // ~200 KB of the 320 KB WGP LDS
  mlp1024_wmma_kernel<<<grid, block, shmem, stream>>>(x, xaug, W1, b1, W2, b2, out);
}